// GPModel_71279277244631
// MI455X (gfx1250) — compile-verified
//
#include <hip/hip_runtime.h>
#include <math.h>

typedef float v2f __attribute__((ext_vector_type(2)));
typedef float v8f __attribute__((ext_vector_type(8)));

#define LDX   8192   // Kxx leading dim
#define NTR   8192
#define NTE   4096
#define NB    128
#define NSTEP (NTR/NB)   // 64

__device__ __forceinline__ v8f wmma4(v2f a, v2f b, v8f c) {
  // D = A(16x4,f32) * B(4x16,f32) + C(16x16,f32)
  return __builtin_amdgcn_wmma_f32_16x16x4_f32(false, a, false, b, (short)0, c,
                                               false, false);
}

// ---------------- feature scaling: out = x / ls, nrm = ||out||^2 -------------
__global__ void k_scale(const float* __restrict__ x, const float* __restrict__ ls,
                        float* __restrict__ out, float* __restrict__ nrm, int n) {
  int i = blockIdx.x * blockDim.x + threadIdx.x;
  if (i >= n) return;
  float s = 0.f;
#pragma unroll
  for (int d = 0; d < 16; ++d) {
    float v = x[i * 16 + d] / ls[d];
    out[i * 16 + d] = v;
    s += v * v;
  }
  nrm[i] = s;
}

// -------- RBF kernel matrix via WMMA gram (K=16): one 16x16 tile / wave ------
__global__ __launch_bounds__(256)
void k_rbf(const float* __restrict__ A, const float* __restrict__ B,
           const float* __restrict__ na, const float* __restrict__ nb,
           float* __restrict__ C, int ldc, int ncolTiles, int nTiles,
           const float* __restrict__ pS, const float* __restrict__ pNoise,
           int addDiagNoise) {
  int wave = (blockIdx.x * blockDim.x + threadIdx.x) >> 5;
  if (wave >= nTiles) return;
  int lane = threadIdx.x & 31;
  int tr = wave / ncolTiles, tc = wave - tr * ncolTiles;
  int r0 = tr * 16, c0 = tc * 16;
  int half = lane >> 4, l15 = lane & 15;
  const float* ap = A + (size_t)(r0 + l15) * 16 + 2 * half;
  const float* bp = B + (size_t)(c0 + l15) * 16 + 2 * half;
  v8f acc = {};
#pragma unroll
  for (int k = 0; k < 16; k += 4) {
    v2f a = *(const v2f*)(ap + k);
    v2f b = *(const v2f*)(bp + k);
    acc = wmma4(a, b, acc);
  }
  float s = pS[0];
  float noise = pNoise[0];
  int col = c0 + l15;
  float nbv = nb[col];
#pragma unroll
  for (int vr = 0; vr < 8; ++vr) {
    int row = r0 + vr + 8 * half;
    float d2 = na[row] + nbv - 2.f * acc[vr];
    d2 = d2 > 0.f ? d2 : 0.f;
    float v = s * __expf(-0.5f * d2);
    if (addDiagNoise && row == col) v += noise;
    C[(size_t)row * ldc + col] = v;
  }
}

// ------------- WMMA GEMM: C (-)= A[MxK] * op(B);  32x32 tile per wave -------
// block = 256 thr = 8 waves (4 row x 2 col) -> 128x64 block tile
// transB==1 : B is [N x K] row-major (use B^T);  else B is [K x N] row-major
// subFromC==1 : C -= A*op(B) ;  else C = A*op(B)
// syrkSkip==1 : skip wave tiles strictly above the diagonal of C
__global__ __launch_bounds__(256)
void k_gemm(float* __restrict__ C, int ldc,
            const float* __restrict__ A, int lda,
            const float* __restrict__ B, int ldb,
            int K, int transB, int subFromC, int syrkSkip) {
  int wave = threadIdx.x >> 5;
  int lane = threadIdx.x & 31;
  int wr = wave >> 1, wc = wave & 1;
  int row0 = blockIdx.x * 128 + wr * 32;
  int col0 = blockIdx.y * 64 + wc * 32;
  if (syrkSkip && col0 > row0 + 31) return;   // tile entirely strictly-upper
  int half = lane >> 4, l15 = lane & 15;

  v8f c00 = {}, c01 = {}, c10 = {}, c11 = {};
  const float* a0p = A + (size_t)(row0 + l15) * lda + 2 * half;
  const float* a1p = a0p + (size_t)16 * lda;

  for (int k = 0; k < K; k += 4) {
    v2f a0 = *(const v2f*)(a0p + k);
    v2f a1 = *(const v2f*)(a1p + k);
    v2f b0, b1;
    if (transB) {
      const float* bp = B + (size_t)(col0 + l15) * ldb + k + 2 * half;
      b0 = *(const v2f*)bp;
      b1 = *(const v2f*)(bp + (size_t)16 * ldb);
    } else {
      int kr = k + 2 * half;
      const float* bp = B + (size_t)kr * ldb + col0 + l15;
      b0.x = bp[0];   b0.y = bp[ldb];
      b1.x = bp[16];  b1.y = bp[ldb + 16];
    }
    c00 = wmma4(a0, b0, c00);
    c01 = wmma4(a0, b1, c01);
    c10 = wmma4(a1, b0, c10);
    c11 = wmma4(a1, b1, c11);
  }

  auto emit = [&](v8f v, int rb, int cb) {
    int col = cb + l15;
#pragma unroll
    for (int vr = 0; vr < 8; ++vr) {
      size_t idx = (size_t)(rb + vr + 8 * half) * ldc + col;
      C[idx] = subFromC ? (C[idx] - v[vr]) : v[vr];
    }
  };
  emit(c00, row0,      col0);
  emit(c01, row0,      col0 + 16);
  emit(c10, row0 + 16, col0);
  emit(c11, row0 + 16, col0 + 16);
}

// --- 128x128 in-LDS Cholesky of diagonal block; also W = inv(L11) to global --
__global__ __launch_bounds__(256)
void k_potf2(float* __restrict__ Ad, int lda, float* __restrict__ W) {
  __shared__ float sA[128 * 128];
  const int tid = threadIdx.x, nt = blockDim.x;
  for (int t = tid; t < 128 * 128; t += nt) {
    sA[t] = Ad[(size_t)(t >> 7) * lda + (t & 127)];
    W[t] = 0.f;                      // zero upper triangle of W
  }
  __syncthreads();
  for (int k = 0; k < 128; ++k) {
    float lkk = sqrtf(sA[k * 128 + k]);
    float inv = 1.f / lkk;
    __syncthreads();
    if (tid == 0) sA[k * 128 + k] = lkk;
    for (int r = k + 1 + tid; r < 128; r += nt) sA[r * 128 + k] *= inv;
    __syncthreads();
    int nn = 127 - k;
    for (int t = tid; t < nn * nn; t += nt) {
      int r = k + 1 + t / nn, c = k + 1 + (t - (t / nn) * nn);
      sA[r * 128 + c] = fmaf(-sA[r * 128 + k], sA[c * 128 + k], sA[r * 128 + c]);
    }
    __syncthreads();
  }
  // forward-substitute L * W = I, one column per thread (lower-triangular W)
  if (tid < 128) {
    const int c = tid;
    W[c * 128 + c] = 1.f / sA[c * 128 + c];
    for (int r = c + 1; r < 128; ++r) {
      float acc = 0.f;
      for (int j = c; j < r; ++j) acc += sA[r * 128 + j] * W[j * 128 + c];
      W[r * 128 + c] = -acc / sA[r * 128 + r];
    }
  }
  __syncthreads();
  for (int t = tid; t < 128 * 128; t += nt)
    Ad[(size_t)(t >> 7) * lda + (t & 127)] = sA[t];
}

// ------------- z = L^{-1} y : single-workgroup forward substitution ----------
__global__ void k_fwdsub_vec(const float* __restrict__ L, int ldl,
                             const float* __restrict__ y, float* __restrict__ z,
                             int n) {
  __shared__ float red[256];
  extern __shared__ float zs[];
  int tid = threadIdx.x;
  for (int i = 0; i < n; ++i) {
    const float* Li = L + (size_t)i * ldl;
    float p = 0.f;
    for (int j = tid; j < i; j += 256) p += Li[j] * zs[j];
    red[tid] = p;
    __syncthreads();
    for (int s = 128; s > 0; s >>= 1) {
      if (tid < s) red[tid] += red[tid + s];
      __syncthreads();
    }
    if (tid == 0) zs[i] = (y[i] - red[0]) / Li[i];
    __syncthreads();
  }
  for (int j = tid; j < n; j += 256) z[j] = zs[j];
}

// ----------------------------- strided 2D copy -------------------------------
__global__ void k_copy(float* __restrict__ dst, int lddst,
                       const float* __restrict__ src, int ldsrc,
                       int rows, int cols) {
  int t = blockIdx.x * blockDim.x + threadIdx.x;
  if (t >= rows * cols) return;
  int r = t / cols, c = t - r * cols;
  dst[(size_t)r * lddst + c] = src[(size_t)r * ldsrc + c];
}

// --------- mean[c] = sum_n V[n,c] z[n] ; var[c] = s + sn - sum_n V^2 ---------
__global__ void k_meanvar(const float* __restrict__ V, const float* __restrict__ z,
                          const float* __restrict__ pS, const float* __restrict__ pN,
                          float* __restrict__ out, int n, int m) {
  int c = blockIdx.x * blockDim.x + threadIdx.x;
  if (c >= m) return;
  float dot = 0.f, ss = 0.f;
  for (int r = 0; r < n; ++r) {
    float v = V[(size_t)r * m + c];
    dot = fmaf(v, z[r], dot);
    ss = fmaf(v, v, ss);
  }
  out[c] = dot;
  out[m + c] = pS[0] + pN[0] - ss;
}

extern "C" void kernel_launch(void* const* d_in, const int* in_sizes, int n_in,
                              void* d_out, int out_size, void* d_ws, size_t ws_size,
                              hipStream_t stream) {
  const float* train_x = (const float*)d_in[0];
  const float* train_y = (const float*)d_in[1];
  const float* test_x  = (const float*)d_in[2];
  const float* pS      = (const float*)d_in[3];
  const float* pLS     = (const float*)d_in[4];
  const float* pNoise  = (const float*)d_in[5];
  float* out = (float*)d_out;

  float* ws   = (float*)d_ws;
  float* Kxx  = ws;                                   // 8192*8192
  float* V    = Kxx + (size_t)NTR * LDX;              // 8192*4096 (Kxs then V)
  float* Wall = V + (size_t)NTR * NTE;                // 64 * 128*128
  float* T    = Wall + (size_t)NSTEP * NB * NB;       // max(8192*128,128*4096)
  float* At   = T + (size_t)NTR * NB;                 // 8192*16
  float* Bt   = At + (size_t)NTR * 16;                // 4096*16
  float* na   = Bt + (size_t)NTE * 16;                // 8192
  float* nb   = na + NTR;                             // 4096
  float* z    = nb + NTE;                             // 8192

  // 1) scaled features + squared norms
  k_scale<<<(NTR + 255) / 256, 256, 0, stream>>>(train_x, pLS, At, na, NTR);
  k_scale<<<(NTE + 255) / 256, 256, 0, stream>>>(test_x,  pLS, Bt, nb, NTE);

  // 2) Kxx = s*exp(-d2/2) + sn*I   (full matrix, WMMA gram)
  {
    int nt = (NTR / 16) * (NTR / 16);
    k_rbf<<<nt / 8, 256, 0, stream>>>(At, At, na, na, Kxx, LDX, NTR / 16, nt,
                                      pS, pNoise, 1);
  }

  // 3) blocked Cholesky (lower), diag-block inverses saved in Wall
  for (int i = 0; i < NSTEP; ++i) {
    float* Ad = Kxx + (size_t)i * NB * LDX + (size_t)i * NB;
    float* Wi = Wall + (size_t)i * NB * NB;
    k_potf2<<<1, 256, 0, stream>>>(Ad, LDX, Wi);
    int nrem = NTR - (i + 1) * NB;
    if (nrem > 0) {
      float* A21 = Kxx + (size_t)(i + 1) * NB * LDX + (size_t)i * NB;
      // panel: T[nrem x 128] = A21 * W^T   (WMMA, K=128)
      dim3 gp(nrem / 128, 2);
      k_gemm<<<gp, 256, 0, stream>>>(T, NB, A21, LDX, Wi, NB, NB, 1, 0, 0);
      k_copy<<<(nrem * NB + 255) / 256, 256, 0, stream>>>(A21, LDX, T, NB,
                                                          nrem, NB);
      // SYRK: A22 -= L21 * L21^T  (lower wave-tiles only)
      float* A22 = Kxx + (size_t)(i + 1) * NB * LDX + (size_t)(i + 1) * NB;
      dim3 gs(nrem / 128, nrem / 64);
      k_gemm<<<gs, 256, 0, stream>>>(A22, LDX, A21, LDX, A21, LDX, NB, 1, 1, 1);
    }
  }

  // 4) z = L^{-1} y
  k_fwdsub_vec<<<1, 256, NTR * sizeof(float), stream>>>(Kxx, LDX, train_y, z, NTR);

  // 5) Kxs into V buffer  [8192 x 4096]
  {
    int nt = (NTR / 16) * (NTE / 16);
    k_rbf<<<nt / 8, 256, 0, stream>>>(At, Bt, na, nb, V, NTE, NTE / 16, nt,
                                      pS, pNoise, 0);
  }

  // 6) V = L^{-1} Kxs, blocked forward TRSM (all WMMA)
  for (int i = 0; i < NSTEP; ++i) {
    float* Bi = V + (size_t)i * NB * NTE;
    if (i > 0) {
      dim3 gu(1, NTE / 64);   // Bi -= L[i, 0:i*128) * V[0:i*128, :]
      k_gemm<<<gu, 256, 0, stream>>>(Bi, NTE, Kxx + (size_t)i * NB * LDX, LDX,
                                     V, NTE, i * NB, 0, 1, 0);
    }
    dim3 gd(1, NTE / 64);     // T = W_i * Bi
    k_gemm<<<gd, 256, 0, stream>>>(T, NTE, Wall + (size_t)i * NB * NB, NB,
                                   Bi, NTE, NB, 0, 0, 0);
    k_copy<<<(NB * NTE + 255) / 256, 256, 0, stream>>>(Bi, NTE, T, NTE, NB, NTE);
  }

  // 7) mean = V^T z ; var = s + sn - colsum(V*V)
  k_meanvar<<<NTE / 256, 256, 0, stream>>>(V, z, pS, pNoise, out, NTR, NTE);
}